// BoundaryHead_73289321939606
// MI455X (gfx1250) — compile-verified
//
#include <hip/hip_runtime.h>
#include <math.h>

#define NB   64
#define NT   2048
#define ND   256
#define NM   20
#define KTOP 100

typedef float v2f __attribute__((ext_vector_type(2)));
typedef float v8f __attribute__((ext_vector_type(8)));

// ---------------------------------------------------------------------------
// Kernel 1: fused projections via V_WMMA_F32_16X16X4_F32.
// One wave computes a 16-row tile of the flattened [B*T] dimension against a
// 4x16 weight tile (cols 0..2 = w_center/w_window/w_offset, rest zero),
// accumulating K=256 in 64 WMMA steps. The weight tile is staged zero-padded
// as [256][16] in LDS so no per-iteration column masking is needed: the inner
// loop is pure global_load_b64 + ds_load + v_wmma. Epilogue applies bias,
// sigmoid*mask for center, and stores the three prediction planes.
__global__ __launch_bounds__(256) void bh_proj_wmma(
    const float* __restrict__ x,
    const float* __restrict__ wc, const float* __restrict__ ww,
    const float* __restrict__ wo,
    const float* __restrict__ bc, const float* __restrict__ bw,
    const float* __restrict__ bo,
    const float* __restrict__ sal,
    float* __restrict__ cent, float* __restrict__ win,
    float* __restrict__ off)
{
    __shared__ float wmat[ND][16];     // zero-padded B tile, 16 KB
    const int tid = threadIdx.x;
    {
        const float w0 = wc[tid];
        const float w1 = ww[tid];
        const float w2 = wo[tid];
#pragma unroll
        for (int j = 0; j < 16; ++j)
            wmat[tid][j] = (j == 0) ? w0 : (j == 1) ? w1 : (j == 2) ? w2 : 0.0f;
    }
    __syncthreads();

    const int lane    = tid & 31;
    const int wave    = (blockIdx.x * blockDim.x + tid) >> 5;   // 8192 waves
    const int rowbase = wave << 4;                              // 16 rows/wave
    const int n       = lane & 15;                              // output column
    const int hi      = lane >> 4;                              // K half select

    const float* __restrict__ xrow = x + (size_t)(rowbase + n) * ND;

    v8f c = {};
#pragma unroll 8
    for (int k0 = 0; k0 < ND; k0 += 4) {
        const int kA = k0 + 2 * hi;
        v2f a, bm;
        a.x  = xrow[kA];
        a.y  = xrow[kA + 1];
        bm.x = wmat[kA][n];
        bm.y = wmat[kA + 1][n];
        c = __builtin_amdgcn_wmma_f32_16x16x4_f32(
                /*neg_a=*/false, a, /*neg_b=*/false, bm,
                /*c_mod=*/(short)0, c, /*reuse_a=*/false, /*reuse_b=*/false);
    }

    // Epilogue: column 0 -> center (sigmoid * mask), 1 -> window, 2 -> offset.
    if (n == 0) {
        const float bias = bc[0];
#pragma unroll
        for (int j = 0; j < 8; ++j) {
            const int row = rowbase + j + 8 * hi;
            const float v = c[j] + bias;
            const float s = 1.0f / (1.0f + __expf(-v));
            const float m = (sal[row] >= 0.0f) ? 1.0f : 0.0f;
            cent[row] = s * m;
        }
    } else if (n == 1) {
        const float bias = bw[0];
#pragma unroll
        for (int j = 0; j < 8; ++j)
            win[rowbase + j + 8 * hi] = c[j] + bias;
    } else if (n == 2) {
        const float bias = bo[0];
#pragma unroll
        for (int j = 0; j < 8; ++j)
            off[rowbase + j + 8 * hi] = c[j] + bias;
    }
}

// ---------------------------------------------------------------------------
// Kernel 2: avg_factor = number of valid objects (b0 != -1).
__global__ __launch_bounds__(256) void bh_avg(
    const float* __restrict__ bnd, float* __restrict__ avg)
{
    __shared__ float red[256];
    const int tid = threadIdx.x;
    float s = 0.0f;
    for (int i = tid; i < NB * NM; i += 256)
        s += (bnd[i * 2] != -1.0f) ? 1.0f : 0.0f;
    red[tid] = s;
    __syncthreads();
    for (int st = 128; st > 0; st >>= 1) {
        if (tid < st) red[tid] += red[tid + st];
        __syncthreads();
    }
    if (tid == 0) avg[0] = red[0];
}

// ---------------------------------------------------------------------------
// Kernel 3: fused target generation + loss partial sums (one block per batch
// row; deterministic per-block partials, no float atomics).
__global__ __launch_bounds__(256) void bh_loss(
    const float* __restrict__ bnd, const float* __restrict__ sal,
    const float* __restrict__ cent, const float* __restrict__ win,
    const float* __restrict__ off, float* __restrict__ partials)
{
    __shared__ int   ovalid[NM], oci[NM], orad[NM];
    __shared__ float ocen[NM], owin[NM], oinv[NM];
    __shared__ float rc[256], rw[256], ro[256];

    const int b   = blockIdx.x;
    const int tid = threadIdx.x;

    if (tid < NM) {
        const float b0 = bnd[(b * NM + tid) * 2 + 0];
        const float b1 = bnd[(b * NM + tid) * 2 + 1];
        const int   v  = (b0 != -1.0f) ? 1 : 0;
        const float b0c = b0 * 0.5f;              // / UNIT
        const float b1c = (b1 - 2.0f) * 0.5f;     // (b1 - UNIT)/UNIT
        const float ctr = fminf((b0c + b1c) * 0.5f, (float)NT - 0.5f);
        const float wnd = b1c - b0c;
        const int   rad = (int)(wnd * 0.2f);      // trunc like .int()
        const float sg  = ((float)rad + 1.0f) * 0.2f;
        ovalid[tid] = v;
        oci[tid]    = (int)ctr;
        orad[tid]   = rad;
        ocen[tid]   = ctr;
        owin[tid]   = wnd;
        oinv[tid]   = 1.0f / (2.0f * sg * sg);
    }
    __syncthreads();

    float cl = 0.0f, wl = 0.0f, ol = 0.0f;
    for (int t = tid; t < NT; t += 256) {
        float gmax = 0.0f, wtg = 0.0f, otg = 0.0f, wgt = 0.0f;
        for (int m = 0; m < NM; ++m) {
            if (!ovalid[m]) continue;
            const int d  = t - oci[m];
            const int ad = (d < 0) ? -d : d;
            if (ad <= orad[m]) {
                const float g = __expf(-(float)(d * d) * oinv[m]);
                gmax = fmaxf(gmax, g);
            }
            if (oci[m] == t) {       // last valid object wins (scatter .set)
                wgt = 1.0f;
                wtg = owin[m];
                otg = ocen[m] - (float)oci[m];
            }
        }
        const int   idx = b * NT + t;
        const float cp  = cent[idx];             // already sigmoid*mask
        const float mk  = (sal[idx] >= 0.0f) ? 1.0f : 0.0f;
        const float pos = (gmax == 1.0f) ? 1.0f : 0.0f;
        const float pl  = -__logf(cp + 1e-12f) * (1.0f - cp) * (1.0f - cp) * pos;
        const float om  = 1.0f - gmax;
        const float om2 = om * om;
        const float nl  = -__logf(1.0f - cp + 1e-12f) * cp * cp * om2 * om2;
        cl += (pl + nl) * mk;
        wl += fabsf(win[idx] - wtg) * wgt;
        ol += fabsf(off[idx] - otg) * wgt;
    }

    rc[tid] = cl; rw[tid] = wl; ro[tid] = ol;
    __syncthreads();
    for (int st = 128; st > 0; st >>= 1) {
        if (tid < st) {
            rc[tid] += rc[tid + st];
            rw[tid] += rw[tid + st];
            ro[tid] += ro[tid + st];
        }
        __syncthreads();
    }
    if (tid == 0) {
        partials[b * 3 + 0] = rc[0];
        partials[b * 3 + 1] = rw[0];
        partials[b * 3 + 2] = ro[0];
    }
}

// ---------------------------------------------------------------------------
// Kernel 4: deterministic loss finalize.
__global__ __launch_bounds__(32) void bh_finalize(
    const float* __restrict__ partials, const float* __restrict__ avg,
    float* __restrict__ losses)
{
    const int tid = threadIdx.x;
    if (tid < 3) {
        float s = 0.0f;
        for (int b = 0; b < NB; ++b) s += partials[b * 3 + tid];
        const float coef = (tid == 1) ? 0.1f : 1.0f;
        losses[tid] = coef * s / avg[0];
    }
}

// ---------------------------------------------------------------------------
// Kernel 5: 3-window peak keep + top-100 (value desc, index asc) per row.
__global__ __launch_bounds__(256) void bh_topk(
    const float* __restrict__ cent, const float* __restrict__ win,
    const float* __restrict__ off, float* __restrict__ outb)
{
    __shared__ float cs[NT];   // staged center_pred row
    __shared__ float cp[NT];   // peak-kept scores (mutated)
    __shared__ float rv[256];
    __shared__ int   ri[256];

    const int b   = blockIdx.x;
    const int tid = threadIdx.x;

    for (int t = tid; t < NT; t += 256) cs[t] = cent[b * NT + t];
    __syncthreads();
    for (int t = tid; t < NT; t += 256) {
        const float c = cs[t];
        const float l = (t > 0)      ? cs[t - 1] : -INFINITY;
        const float r = (t < NT - 1) ? cs[t + 1] : -INFINITY;
        cp[t] = (c >= l && c >= r) ? c : 0.0f;
    }
    __syncthreads();

    for (int k = 0; k < KTOP; ++k) {
        float bv = -INFINITY;
        int   bi = NT;
        for (int t = tid; t < NT; t += 256) {
            const float v = cp[t];
            if (v > bv || (v == bv && t < bi)) { bv = v; bi = t; }
        }
        rv[tid] = bv; ri[tid] = bi;
        __syncthreads();
        for (int st = 128; st > 0; st >>= 1) {
            if (tid < st) {
                const float v2 = rv[tid + st];
                const int   i2 = ri[tid + st];
                if (v2 > rv[tid] || (v2 == rv[tid] && i2 < ri[tid])) {
                    rv[tid] = v2; ri[tid] = i2;
                }
            }
            __syncthreads();
        }
        if (tid == 0) {
            const int   idx   = ri[0];
            const float score = rv[0];
            cp[idx] = -1.0f;                       // all candidates are >= 0
            const float ov = fmaxf(off[b * NT + idx], 0.0f);
            const float wv = fmaxf(win[b * NT + idx], 0.0f);
            const float cc = (float)idx + ov;
            const float lo = fminf(fmaxf(cc - wv * 0.5f, 0.0f),
                                   (float)(NT - 1)) * 2.0f;
            const float hi = fminf(fmaxf(cc + wv * 0.5f, 0.0f),
                                   (float)(NT - 1)) * 2.0f + 2.0f;
            float* o = outb + (size_t)(b * KTOP + k) * 3;
            o[0] = lo; o[1] = hi; o[2] = score;
        }
        __syncthreads();
    }
}

// ---------------------------------------------------------------------------
extern "C" void kernel_launch(void* const* d_in, const int* in_sizes, int n_in,
                              void* d_out, int out_size, void* d_ws,
                              size_t ws_size, hipStream_t stream)
{
    const float* x   = (const float*)d_in[0];
    const float* wc  = (const float*)d_in[1];
    const float* bc  = (const float*)d_in[2];
    const float* ww  = (const float*)d_in[3];
    const float* bw  = (const float*)d_in[4];
    const float* wo  = (const float*)d_in[5];
    const float* bo  = (const float*)d_in[6];
    const float* sal = (const float*)d_in[7];
    const float* bnd = (const float*)d_in[8];

    float* out      = (float*)d_out;                 // [64,100,3] + 3 losses
    float* cent     = (float*)d_ws;                  // 131072
    float* win      = cent + NB * NT;                // 131072
    float* off      = win + NB * NT;                 // 131072
    float* avg      = off + NB * NT;                 // 1
    float* partials = avg + 1;                       // 192

    bh_proj_wmma<<<1024, 256, 0, stream>>>(x, wc, ww, wo, bc, bw, bo, sal,
                                           cent, win, off);
    bh_avg<<<1, 256, 0, stream>>>(bnd, avg);
    bh_loss<<<NB, 256, 0, stream>>>(bnd, sal, cent, win, off, partials);
    bh_finalize<<<1, 32, 0, stream>>>(partials, avg, out + NB * KTOP * 3);
    bh_topk<<<NB, 256, 0, stream>>>(cent, win, off, out);
}